// ResidueFeatureAggregator_73229192397400
// MI455X (gfx1250) — compile-verified
//
#include <hip/hip_runtime.h>

typedef float v2f __attribute__((ext_vector_type(2)));
typedef float v4f __attribute__((ext_vector_type(4)));
typedef float v8f __attribute__((ext_vector_type(8)));

#define ATOM_DIM 128
#define HIDDEN   256
#define RES_DIM  20
#define TILE_M   16
#define LDA      132   // padded LDS stride for A tile (132 % 64 == 4 -> conflict-free col reads)
#define LDH      260   // padded LDS stride for H tile (260 % 64 == 4)

__device__ __forceinline__ int lower_bound_i32(const int* __restrict__ m, int n, int v) {
    int lo = 0, hi = n;
    while (lo < hi) {
        int mid = (lo + hi) >> 1;
        if (m[mid] < v) lo = mid + 1; else hi = mid;
    }
    return lo;
}

__global__ __launch_bounds__(256)
void residue_mlp_fused(const float* __restrict__ atom_features,
                       const int*   __restrict__ residue_mapping,
                       const float* __restrict__ W1, const float* __restrict__ bias1,
                       const float* __restrict__ W2, const float* __restrict__ bias2,
                       float* __restrict__ out,
                       int num_atoms, int num_res)
{
    __shared__ float a_lds[TILE_M * LDA];   // 16 x 128 residue means (padded)
    __shared__ float h_lds[TILE_M * LDH];   // 16 x 256 hidden activations (padded)
    __shared__ int   bounds[TILE_M + 1];

    const int tid  = threadIdx.x;
    const int wave = tid >> 5;
    const int lane = tid & 31;
    const int m0   = blockIdx.x * TILE_M;

    // ---- Phase 0: residue boundaries via parallel binary searches (sorted mapping) ----
    if (tid <= TILE_M) {
        int v = m0 + tid; if (v > num_res) v = num_res;
        bounds[tid] = lower_bound_i32(residue_mapping, num_atoms, v);
    }
    __syncthreads();

    // ---- Phase 1: segment mean, 2 residues per wave; lane owns 4 contiguous features ----
    for (int i = 0; i < 2; ++i) {
        const int rl = wave * 2 + i;                 // local residue row 0..15
        const int s = bounds[rl], e = bounds[rl + 1];
        v4f acc = {0.f, 0.f, 0.f, 0.f};
        const float* base = atom_features + (size_t)s * ATOM_DIM + lane * 4;
        for (int a = s; a < e; ++a) {
            v4f v = __builtin_nontemporal_load((const v4f*)base);   // streamed, read-once
            acc += v;
            base += ATOM_DIM;
        }
        const float inv = (e > s) ? 1.f / (float)(e - s) : 0.f;
        acc *= inv;
        *(v4f*)&a_lds[rl * LDA + lane * 4] = acc;
    }
    __syncthreads();

    // WMMA lane geometry (f32 16x16x4): A lanes 0-15 -> K {k,k+1}, lanes 16-31 -> K {k+2,k+3}
    const int arow = lane & 15;
    const int bcol = lane & 15;
    const int koff = (lane < 16) ? 0 : 2;
    const int hrow = (lane < 16) ? 0 : 8;

    // ---- Phase 2: GEMM1  [16x128] @ [128x256]  (wave w owns columns [32w, 32w+32)) ----
    const int n0 = wave * 32;
    v8f c0 = {}; v8f c1 = {};
    #pragma unroll 4
    for (int kk = 0; kk < ATOM_DIM; kk += 4) {
        v2f a;
        a.x = a_lds[arow * LDA + kk + koff];
        a.y = a_lds[arow * LDA + kk + koff + 1];
        const float* w1p = W1 + (size_t)(kk + koff) * HIDDEN + bcol;
        v2f bA, bB;
        bA.x = w1p[n0];               bA.y = w1p[HIDDEN + n0];
        bB.x = w1p[n0 + 16];          bB.y = w1p[HIDDEN + n0 + 16];
        c0 = __builtin_amdgcn_wmma_f32_16x16x4_f32(false, a, false, bA, (short)0, c0, false, false);
        c1 = __builtin_amdgcn_wmma_f32_16x16x4_f32(false, a, false, bB, (short)0, c1, false, false);
    }

    // bias + leaky_relu(0.01), scatter to LDS hidden tile
    {
        const float g0 = bias1[n0 + bcol];
        const float g1 = bias1[n0 + 16 + bcol];
        #pragma unroll
        for (int j = 0; j < 8; ++j) {
            float v0 = c0[j] + g0;
            float v1 = c1[j] + g1;
            v0 = v0 > 0.f ? v0 : 0.01f * v0;
            v1 = v1 > 0.f ? v1 : 0.01f * v1;
            h_lds[(hrow + j) * LDH + n0 + bcol]      = v0;
            h_lds[(hrow + j) * LDH + n0 + 16 + bcol] = v1;
        }
    }
    __syncthreads();

    // ---- Phase 3: GEMM2  [16x256] @ [256x20]  (waves 0,1 cover N padded to 32) ----
    if (wave < 2) {
        const int n0b = wave * 16;
        const int col = n0b + bcol;
        const bool okc = col < RES_DIM;
        v8f c = {};
        #pragma unroll 4
        for (int kk = 0; kk < HIDDEN; kk += 4) {
            v2f a;
            a.x = h_lds[arow * LDH + kk + koff];
            a.y = h_lds[arow * LDH + kk + koff + 1];
            v2f b;
            b.x = okc ? W2[(size_t)(kk + koff) * RES_DIM + col]     : 0.f;
            b.y = okc ? W2[(size_t)(kk + koff + 1) * RES_DIM + col] : 0.f;
            c = __builtin_amdgcn_wmma_f32_16x16x4_f32(false, a, false, b, (short)0, c, false, false);
        }
        if (okc) {
            const float g = bias2[col];
            #pragma unroll
            for (int j = 0; j < 8; ++j) {
                const int row = m0 + hrow + j;
                if (row < num_res) out[(size_t)row * RES_DIM + col] = c[j] + g;
            }
        }
    }
}

extern "C" void kernel_launch(void* const* d_in, const int* in_sizes, int n_in,
                              void* d_out, int out_size, void* d_ws, size_t ws_size,
                              hipStream_t stream) {
    const float* atom = (const float*)d_in[0];
    const int*   map  = (const int*)d_in[1];
    const float* W1   = (const float*)d_in[2];
    const float* b1   = (const float*)d_in[3];
    const float* W2   = (const float*)d_in[4];
    const float* b2   = (const float*)d_in[5];
    float* out = (float*)d_out;

    const int num_atoms = in_sizes[1];            // residue_mapping length
    const int num_res   = out_size / RES_DIM;     // 125000
    const int nblocks   = (num_res + TILE_M - 1) / TILE_M;

    residue_mlp_fused<<<nblocks, 256, 0, stream>>>(atom, map, W1, b1, W2, b2, out,
                                                   num_atoms, num_res);
}